// ClosedFlyLoop_74715251081390
// MI455X (gfx1250) — compile-verified
//
#include <hip/hip_runtime.h>
#include <math.h>

// ---------------------------------------------------------------------------
// Problem geometry (fixed by the reference's setup_inputs)
// ---------------------------------------------------------------------------
#define H_IMG 1024
#define W_IMG 2048
#define HW (H_IMG * W_IMG)
#define RAD 20              // RADIUS = 4*SIGMA + 0.5 with SIGMA=5
#define KLEN 41
#define SPAN 56             // 16 outputs + 2*RAD halo
#define TILE_COLS 104       // 64 outputs + 2*RAD halo
#define LDS_STRIDE 105      // +1 pad: 105 mod 64 = 41, coprime -> conflict-free column reads
#define INV_D (1.0f / 2.27f)
#define NRED_BLOCKS 1024

typedef __attribute__((ext_vector_type(2))) float v2f;
typedef __attribute__((ext_vector_type(8))) float v8f;

// D(16x16,f32) += A(16x4,f32) x B(4x16,f32)  — V_WMMA_F32_16X16X4_F32
__device__ __forceinline__ v8f wmma_k4(v2f a, v2f b, v8f c) {
  return __builtin_amdgcn_wmma_f32_16x16x4_f32(
      /*neg_a=*/false, a, /*neg_b=*/false, b,
      /*c_mod=*/(short)0, c, /*reuse_a=*/false, /*reuse_b=*/false);
}

// ---------------------------------------------------------------------------
// Pass 1: deterministic reduction  sum_pixels sqrt(m00^2+m01^2+m10^2+m11^2)
// Fixed block/thread assignment + fixed-order tree => bit-identical replays.
// ---------------------------------------------------------------------------
__global__ __launch_bounds__(256) void reduce_stage1(const float* __restrict__ y,
                                                     float* __restrict__ part) {
  float s = 0.f;
  for (int i = blockIdx.x * 256 + threadIdx.x; i < HW; i += NRED_BLOCKS * 256) {
    float a = y[i], b = y[HW + i], c = y[2 * HW + i], d = y[3 * HW + i];
    s += sqrtf(a * a + b * b + c * c + d * d);
  }
  __shared__ float red[256];
  red[threadIdx.x] = s;
  __syncthreads();
  for (int off = 128; off > 0; off >>= 1) {
    if (threadIdx.x < off) red[threadIdx.x] += red[threadIdx.x + off];
    __syncthreads();
  }
  if (threadIdx.x == 0) part[blockIdx.x] = red[0];
}

__global__ __launch_bounds__(256) void reduce_stage2(float* __restrict__ part) {
  __shared__ float red[256];
  int t = threadIdx.x;
  float s = part[t] + part[t + 256] + part[t + 512] + part[t + 768];
  red[t] = s;
  __syncthreads();
  for (int off = 128; off > 0; off >>= 1) {
    if (t < off) red[t] += red[t + off];
    __syncthreads();
  }
  if (t == 0) part[NRED_BLOCKS] = red[0];   // total
}

// ---------------------------------------------------------------------------
// Pass 2: fused Toeplitz-WMMA convolutions + per-pixel tensor algebra.
// Block = 128 threads (4 wave32), tile = 16 rows x 64 cols. Wave w owns the
// 16x16 subtile at column offset 16*w. For each of the 7 channels the block
// loads one (56 x 104) halo tile into LDS (wrap in Y, reflect in X), then:
//   dY(16x16) = Toep(16x56) * S(56x16)      [K chunked by 4, 14 WMMA]
//   dX(16x16) = S(16x56)    * Toep^T(56x16) [14 WMMA]
// Toeplitz: T[a,b] = K[b-a] (K indexed 0..40 == offset -20..20), zero off-band,
// with the 1/2.27 output scale folded into the taps (linear in both products).
// The Toeplitz fragments depend only on (chunk, lane): hoisted into registers
// once, reused across all 7 channels -> steady-state inner loop is
// 2 ds_load + 2 v_wmma per chunk.
// Per-lane fragment values of Toep-as-A (m=lane&15) and Toep^T-as-B
// (n=lane&15) coincide, so one fragment serves both multiplies.
// Accumulator layout (C/D 16x16 f32): lane&15 = N, vgpr e (+8 if lane>=16) = M,
// which directly gives each lane its 8 owned pixels for the pointwise stage.
// ---------------------------------------------------------------------------
__global__ __launch_bounds__(128) void fused_kernel(
    const float* __restrict__ y, const float* __restrict__ v,
    const float* __restrict__ gamma, const float* __restrict__ cadc,
    const float* __restrict__ myoc, const float* __restrict__ wsum,
    float* __restrict__ out) {
  __shared__ float tile[SPAN][LDS_STRIDE];
  __shared__ float kt[44];

  const int tid = threadIdx.x;

  // Gaussian first-derivative taps: k[i] = -(x/sigma^2)*phi(x)/sum(phi) * INV_D
  if (tid < KLEN) {
    float x = (float)(tid - RAD);
    kt[tid] = expf(-0.5f * x * x / 25.f);
  }
  __syncthreads();
  if (tid == 0) {
    float s = 0.f;
    for (int i = 0; i < KLEN; ++i) s += kt[i];
    kt[43] = s;
  }
  __syncthreads();
  if (tid < KLEN) {
    float x = (float)(tid - RAD);
    kt[tid] = (-x / 25.f) * kt[tid] / kt[43] * INV_D;
  }
  __syncthreads();

  const int x0 = blockIdx.x * 64;
  const int y0 = blockIdx.y * 16;
  const int lane = tid & 31;
  const int wv = tid >> 5;      // wave id 0..3
  const int lr = lane & 15;
  const int hi = lane >> 4;     // 0: K pair {0,1}; 1: K pair {2,3}
  const int xb = wv * 16;       // subtile column offset in LDS-output space

  // Hoisted banded-Toeplitz fragments: channel-invariant, (chunk,lane)-only.
  // Unconditional clamped LDS read + select: no EXEC change near WMMA.
  v2f tf[14];
#pragma unroll
  for (int i = 0; i < 14; ++i) {
    const int k0 = 4 * i + 2 * hi;
    int d0 = k0 - lr, d1 = d0 + 1;
    int c0 = min(max(d0, 0), 40), c1 = min(max(d1, 0), 40);
    float t0 = kt[c0]; t0 = (d0 == c0) ? t0 : 0.f;
    float t1 = kt[c1]; t1 = (d1 == c1) ? t1 : 0.f;
    tf[i][0] = t0;
    tf[i][1] = t1;
  }

  v8f dYr[7], dXr[7];

#pragma unroll
  for (int ch = 0; ch < 7; ++ch) {
    __syncthreads();  // previous iteration's LDS consumers done
    const float* src = (ch < 5) ? (y + ch * HW) : (v + (ch - 5) * HW);
    for (int idx = tid; idx < SPAN * TILE_COLS; idx += 128) {
      int r = idx / TILE_COLS, c = idx - r * TILE_COLS;
      int gy = y0 - RAD + r;                 // wrap (diffY uses mode='wrap')
      gy += (gy < 0) ? H_IMG : 0;
      gy -= (gy >= H_IMG) ? H_IMG : 0;
      int gx = x0 - RAD + c;                 // reflect (diffX uses 'reflect')
      gx = (gx < 0) ? -gx : gx;
      gx = (gx >= W_IMG) ? (2 * W_IMG - 2 - gx) : gx;
      tile[r][c] = src[gy * W_IMG + gx];
    }
    __syncthreads();

    v8f accY = {0.f, 0.f, 0.f, 0.f, 0.f, 0.f, 0.f, 0.f};
    v8f accX = {0.f, 0.f, 0.f, 0.f, 0.f, 0.f, 0.f, 0.f};
#pragma unroll
    for (int i = 0; i < 14; ++i) {
      const int k0 = 4 * i + 2 * hi;         // absolute K for vgpr 0; +1 for vgpr 1
      // Y-derivative: A=Toep, B=input columns  S[r][n] = tile[r][xb+20+n]
      v2f sb = {tile[k0][xb + RAD + lr], tile[k0 + 1][xb + RAD + lr]};
      accY = wmma_k4(tf[i], sb, accY);
      // X-derivative: A=input rows S[m][q] = tile[m+20][xb+q], B=Toep^T
      v2f sa = {tile[lr + RAD][xb + k0], tile[lr + RAD][xb + k0 + 1]};
      accX = wmma_k4(sa, tf[i], accX);
    }
    dYr[ch] = accY;   // 1/2.27 already folded into taps
    dXr[ch] = accX;
  }

  // ---------------- pointwise tensor algebra ----------------
  const float m0bar = wsum[0] * (1.0f / (float)HW);
  const float cad0 = fmaxf(cadc[0], 0.f), cad1 = fmaxf(cadc[1], 0.f),
              cad2 = fmaxf(cadc[2], 0.f);
  const float my0 = fmaxf(myoc[0], 0.f), my1 = fmaxf(myoc[1], 0.f),
              my2 = fmaxf(myoc[2], 0.f), my3 = fmaxf(myoc[3], 0.f),
              my4 = fmaxf(myoc[4], 0.f);

#pragma unroll
  for (int e = 0; e < 8; ++e) {
    const int row = y0 + e + 8 * hi;
    const int col = x0 + xb + lr;
    const int p = row * W_IMG + col;

    const float m00 = y[p], m01 = y[HW + p], m10 = y[2 * HW + p], m11 = y[3 * HW + p];
    const float cc = y[4 * HW + p];
    const float vv0 = v[p], vv1 = v[HW + p];
    const float g = gamma[p];

    const float dYv0 = dYr[5][e], dXv0 = dXr[5][e];
    const float dYv1 = dYr[6][e], dXv1 = dXr[6][e];

    const float o = -0.5f * (dXv0 - dYv1);                 // O[0][1]
    const float E00 = dYv0, E11 = dXv1, E01 = 0.5f * (dXv0 + dYv1);
    const float trm = m00 + m11;
    const float dev00 = m00 - 0.5f * trm, dev11 = m11 - 0.5f * trm;
    const float dev01 = m01, dev10 = m10;
    const float dmag2 = dev00 * dev00 + dev01 * dev01 + dev10 * dev10 + dev11 * dev11;
    const float devE = dev00 * E00 + dev11 * E11 + (dev01 + dev10) * E01;
    const float adE = fabsf(devE);                         // sign(devE)*devE
    const float sc = 0.5f * sqrtf(dmag2) / m0bar;
    const float Ea00 = (E00 - adE * dev00 / dmag2) * sc;
    const float Ea01 = (E01 - adE * dev01 / dmag2) * sc;
    const float Ea10 = (E01 - adE * dev10 / dmag2) * sc;
    const float Ea11 = (E11 - adE * dev11 / dmag2) * sc;
    const float Ep00 = E00 - Ea00, Ep01 = E01 - Ea01;
    const float Ep10 = E01 - Ea10, Ep11 = E11 - Ea11;
    // mE = m*Ep + Ep*m
    const float mE00 = (m00 * Ep00 + m01 * Ep10) + (Ep00 * m00 + Ep01 * m10);
    const float mE01 = (m00 * Ep01 + m01 * Ep11) + (Ep00 * m01 + Ep01 * m11);
    const float mE10 = (m10 * Ep00 + m11 * Ep10) + (Ep10 * m00 + Ep11 * m10);
    const float mE11 = (m10 * Ep01 + m11 * Ep11) + (Ep10 * m01 + Ep11 * m11);
    // -(O*m) + (m*O) with O = [[0,o],[-o,0]]
    const float rot00 = -o * m10 - o * m01;
    const float rot01 = -o * m11 + o * m00;
    const float rot10 =  o * m00 - o * m11;
    const float rot11 =  o * m01 + o * m10;

    const float cf = my1 - my2 * cc;
    const float md00 = -(vv0 * dYr[0][e] + vv1 * dXr[0][e]) + rot00 - my0 * m00 +
                       cf * mE00 + my3 * trm + my4 * trm * m00;   // gamma_dv[0][0]=1
    const float md01 = -(vv0 * dYr[1][e] + vv1 * dXr[1][e]) + rot01 - my0 * m01 +
                       cf * mE01 + my4 * trm * m01;
    const float md10 = -(vv0 * dYr[2][e] + vv1 * dXr[2][e]) + rot10 - my0 * m10 +
                       cf * mE10 + my4 * trm * m10;
    const float md11 = -(vv0 * dYr[3][e] + vv1 * dXr[3][e]) + rot11 - my0 * m11 +
                       cf * mE11 + my4 * trm * m11;
    const float cd = -(vv0 * dYr[4][e] + vv1 * dXr[4][e]) - cad0 * cc +
                     cad1 * cc * (dYv0 + dXv1) + cad2 * g;

    out[p] = md00;
    out[HW + p] = md01;
    out[2 * HW + p] = md10;
    out[3 * HW + p] = md11;
    out[4 * HW + p] = cd;
  }
}

// ---------------------------------------------------------------------------
extern "C" void kernel_launch(void* const* d_in, const int* in_sizes, int n_in,
                              void* d_out, int out_size, void* d_ws, size_t ws_size,
                              hipStream_t stream) {
  const float* y = (const float*)d_in[0];      // (5,H,W)
  const float* v = (const float*)d_in[1];      // (2,H,W)
  const float* gamma = (const float*)d_in[2];  // (H,W)
  const float* cad = (const float*)d_in[3];    // (3,)
  const float* myo = (const float*)d_in[4];    // (5,)
  float* out = (float*)d_out;                  // (5,H,W)
  float* ws = (float*)d_ws;                    // ws[0..1023]=partials, ws[1024]=sum

  reduce_stage1<<<NRED_BLOCKS, 256, 0, stream>>>(y, ws);
  reduce_stage2<<<1, 256, 0, stream>>>(ws);
  dim3 grid(W_IMG / 64, H_IMG / 16);
  fused_kernel<<<grid, 128, 0, stream>>>(y, v, gamma, cad, myo, ws + NRED_BLOCKS, out);
}